// MaskedAutoregressiveFlow_4681514352656
// MI455X (gfx1250) — compile-verified
//
#include <hip/hip_runtime.h>
#include <hip/hip_bf16.h>

#define DIM 64
#define HID 512
#define BATCH 8192

typedef __attribute__((ext_vector_type(2))) float v2f;
typedef __attribute__((ext_vector_type(8))) float v8f;

// Workspace layout (floats):
//   [0, 32768)        : MW1T (64 x 512), MW1T[k*512+j] = mask1[j,k]*W1[j,k]   (scan)
//   [32768, 98304)    : MW2  (128 x 512) row-major                            (scan)
//   [98304, 98816)    : w_sum[j] = sum_{r=64..127} MW2[r,j]
//   [98816]           : b_sum = sum_{r=64..127} b2[r]
//   [98832, 131600)   : packedB: MW1^T pre-swizzled into WMMA 16x4 f32 lane layout
//                       pair index p = (nt*16 + t)*32 + lane, holds
//                       { B[4t+koff][nt*16+lr], B[4t+koff+1][nt*16+lr] },
//                       koff = (lane>=16)*2, lr = lane&15
#define WS_MW1T  0
#define WS_MW2   (HID*DIM)
#define WS_WSUM  (HID*DIM + 2*DIM*HID)
#define WS_BSUM  (WS_WSUM + HID)
#define WS_BPACK 98832
#define NPACK    (HID*DIM)            // 32768 floats

// ---------------- prep: apply masks, build transposed + WMMA-packed W1, masked W2 ----------
__global__ __launch_bounds__(256) void maf_prep_mask(const float* __restrict__ W1,
                                                     const float* __restrict__ mask1,
                                                     const float* __restrict__ W2,
                                                     const float* __restrict__ mask2,
                                                     float* __restrict__ ws) {
  int idx = blockIdx.x * blockDim.x + threadIdx.x;
  if (idx < HID * DIM) {
    int j = idx / DIM, k = idx % DIM;                 // W1 is (512,64) row-major
    ws[WS_MW1T + k * HID + j] = mask1[idx] * W1[idx]; // store transposed (64x512)
  } else if (idx < HID * DIM + 2 * DIM * HID) {
    int t = idx - HID * DIM;                          // W2 is (128,512) row-major
    ws[WS_MW2 + t] = mask2[t] * W2[t];
  } else if (idx < HID * DIM + 2 * DIM * HID + NPACK) {
    int f = idx - (HID * DIM + 2 * DIM * HID);        // packed-B float index
    int p = f >> 1, c = f & 1;                        // pair, element-in-pair
    int nt = p >> 9;                                  // / (16*32)
    int t  = (p >> 5) & 15;
    int l  = p & 31;
    int koff = (l >> 4) << 1;
    int j = (nt << 4) + (l & 15);                     // hidden index (N)
    int k = (t << 2) + koff + c;                      // input index (K)
    ws[WS_BPACK + f] = mask1[j * DIM + k] * W1[j * DIM + k];
  }
}

__global__ __launch_bounds__(512) void maf_prep_sum(const float* __restrict__ b2,
                                                    float* __restrict__ ws) {
  int j = threadIdx.x;                                // 512 threads
  const float* MW2 = ws + WS_MW2;
  float s = 0.f;
  for (int r = DIM; r < 2 * DIM; ++r) s += MW2[r * HID + j];
  ws[WS_WSUM + j] = s;
  if (j == 0) {
    float bs = 0.f;
    for (int r = DIM; r < 2 * DIM; ++r) bs += b2[r];
    ws[WS_BSUM] = bs;
  }
}

// ---------------- scan: one wave32 per batch row, h_pre resident in VGPRs ----------------
__global__ __launch_bounds__(256) void maf_scan(const float* __restrict__ z,
                                                const float* __restrict__ b1,
                                                const float* __restrict__ b2,
                                                const float* __restrict__ ws,
                                                float* __restrict__ xout) {
  __shared__ __align__(16) float w1c[HID];   // MW1T row i  (column i of masked W1)
  __shared__ __align__(16) float w2a[HID];   // MW2 row i      -> mu weights
  __shared__ __align__(16) float w2b[HID];   // MW2 row 64+i   -> alpha weights
  const float* MW1T = ws + WS_MW1T;
  const float* MW2  = ws + WS_MW2;

  const int tid  = threadIdx.x;
  const int lane = tid & 31;
  const int wave = tid >> 5;
  const int row  = blockIdx.x * 8 + wave;   // 1024 blocks * 8 waves = 8192 rows
  const int t2   = tid * 2;                 // consecutive pair per thread -> b64 ld/st

  // h_pre for this row: 512 f32 striped as h[t] <-> hidden index lane + 32*t
  float h[16];
#pragma unroll
  for (int t = 0; t < 16; ++t) h[t] = b1[lane + 32 * t];

  const float z0 = z[row * DIM + lane];        // cols 0..31
  const float z1 = z[row * DIM + 32 + lane];   // cols 32..63
  float x0 = 0.f, x1 = 0.f;

  for (int i = 0; i < DIM; ++i) {
    __syncthreads();  // previous iteration's LDS reads complete
    *(v2f*)&w1c[t2] = *(const v2f*)&MW1T[i * HID + t2];
    *(v2f*)&w2a[t2] = *(const v2f*)&MW2[i * HID + t2];
    *(v2f*)&w2b[t2] = *(const v2f*)&MW2[(DIM + i) * HID + t2];
    __syncthreads();

    float mu = 0.f, al = 0.f;
#pragma unroll
    for (int t = 0; t < 16; ++t) {
      float hv = fmaxf(h[t], 0.f);           // relu at read; keep pre-activation
      int j = lane + 32 * t;
      mu = fmaf(hv, w2a[j], mu);
      al = fmaf(hv, w2b[j], al);
    }
    // butterfly reduce over the wave: all 32 lanes end with the full sums
#pragma unroll
    for (int off = 16; off >= 1; off >>= 1) {
      mu += __shfl_xor(mu, off, 32);
      al += __shfl_xor(al, off, 32);
    }
    mu += b2[i];
    al += b2[DIM + i];

    float zsel = (i < 32) ? z0 : z1;
    float zi = __shfl(zsel, i & 31, 32);
    float xi = fmaf(zi, expf(al), mu);       // x[:,i] = z[:,i]*exp(alpha)+mu

    if (i < 32) { if (lane == i)        x0 = xi; }
    else        { if (lane == (i - 32)) x1 = xi; }

    if (i < DIM - 1) {                       // rank-1 update of h_pre with new column
#pragma unroll
      for (int t = 0; t < 16; ++t) h[t] = fmaf(xi, w1c[lane + 32 * t], h[t]);
    }
  }

  xout[row * DIM + lane]      = x0;
  xout[row * DIM + 32 + lane] = x1;
}

// ---------------- final: log_det via fp32 WMMA GEMM fused with bias/relu/dot ----------------
__global__ __launch_bounds__(256) void maf_logdet_wmma(const float* __restrict__ b1,
                                                       const float* __restrict__ ws,
                                                       const float* __restrict__ xout,
                                                       float* __restrict__ ldout) {
  __shared__ float b1s[HID];
  __shared__ float wss[HID];
  const v2f* Bp = (const v2f*)(ws + WS_BPACK);   // WMMA-lane-packed MW1^T
  const float bsum = ws[WS_BSUM];

  const int tid = threadIdx.x;
  b1s[tid]       = b1[tid];
  b1s[tid + 256] = b1[tid + 256];
  wss[tid]       = ws[WS_WSUM + tid];
  wss[tid + 256] = ws[WS_WSUM + tid + 256];
  __syncthreads();

  const int lane = tid & 31;
  const int wave = tid >> 5;
  const int lr   = lane & 15;
  const int koff = (lane >> 4) ? 2 : 0;    // f32 A/B tiles: low lanes hold K 0,1; high lanes K 2,3
  const int m0   = (blockIdx.x * 8 + wave) * 16;   // 64 blocks * 8 waves * 16 rows = 8192

  // Preload A tiles: X(16x64) as 16 tiles of 16x4 f32; per-lane pair is contiguous -> b64 loads
  const float* xp = xout + (m0 + lr) * DIM + koff;
  v2f a[16];
#pragma unroll
  for (int t = 0; t < 16; ++t) a[t] = *(const v2f*)(xp + 4 * t);

  v8f ld = {};   // per-lane partial of log_det, row = vgpr (+8 in upper half-wave)
  for (int nt = 0; nt < 32; ++nt) {
    const int col = nt * 16 + lr;          // hidden index (N) handled by this lane

    // clause-friendly: fetch all 16 packed B tiles (1x b64 per lane per tile)
    v2f bt[16];
#pragma unroll
    for (int t = 0; t < 16; ++t) bt[t] = Bp[(nt * 16 + t) * 32 + lane];

    // two accumulator chains for WMMA ILP
    v8f acc0 = {}, acc1 = {};
#pragma unroll
    for (int t = 0; t < 16; t += 2) {
      acc0 = __builtin_amdgcn_wmma_f32_16x16x4_f32(false, a[t],     false, bt[t],
                                                   (short)0, acc0, false, false);
      acc1 = __builtin_amdgcn_wmma_f32_16x16x4_f32(false, a[t + 1], false, bt[t + 1],
                                                   (short)0, acc1, false, false);
    }
    v8f acc = acc0 + acc1;

    const float bb = b1s[col];
    const float wv = wss[col];
#pragma unroll
    for (int v = 0; v < 8; ++v) {
      float hv = fmaxf(acc[v] + bb, 0.f);  // h = relu(x*MW1^T + b1)
      ld[v] = fmaf(hv, wv, ld[v]);         // accumulate h . w_sum
    }
  }

  // reduce across the 16 lanes of each half-wave (xor 1,2,4,8 stays within the half)
#pragma unroll
  for (int off = 1; off < 16; off <<= 1) {
#pragma unroll
    for (int v = 0; v < 8; ++v) {
      float t = ld[v];
      ld[v] = t + __shfl_xor(t, off, 32);
    }
  }
  if (lane == 0) {
#pragma unroll
    for (int v = 0; v < 8; ++v) ldout[m0 + v] = ld[v] + bsum;        // rows m0..m0+7
  } else if (lane == 16) {
#pragma unroll
    for (int v = 0; v < 8; ++v) ldout[m0 + 8 + v] = ld[v] + bsum;    // rows m0+8..m0+15
  }
}

extern "C" void kernel_launch(void* const* d_in, const int* in_sizes, int n_in,
                              void* d_out, int out_size, void* d_ws, size_t ws_size,
                              hipStream_t stream) {
  const float* z     = (const float*)d_in[0];   // (8192, 64)
  const float* W1    = (const float*)d_in[1];   // (512, 64)
  const float* b1    = (const float*)d_in[2];   // (512,)
  const float* W2    = (const float*)d_in[3];   // (128, 512)
  const float* b2    = (const float*)d_in[4];   // (128,)
  const float* mask1 = (const float*)d_in[5];   // (512, 64)
  const float* mask2 = (const float*)d_in[6];   // (128, 512)

  float* ws    = (float*)d_ws;
  float* xout  = (float*)d_out;                 // (8192, 64)
  float* ldout = xout + BATCH * DIM;            // (8192,)

  const int nprep = HID * DIM + 2 * DIM * HID + NPACK;  // 131072
  maf_prep_mask<<<(nprep + 255) / 256, 256, 0, stream>>>(W1, mask1, W2, mask2, ws);
  maf_prep_sum<<<1, 512, 0, stream>>>(b2, ws);
  maf_scan<<<BATCH / 8, 256, 0, stream>>>(z, b1, b2, ws, xout);
  maf_logdet_wmma<<<BATCH / 128, 256, 0, stream>>>(b1, ws, xout, ldout);
}